// CompositePerturbation_16106127360110
// MI455X (gfx1250) — compile-verified
//
#include <hip/hip_runtime.h>
#include <hip/hip_bf16.h>

typedef __attribute__((ext_vector_type(2))) float v2f;
typedef __attribute__((ext_vector_type(8))) float v8f;

#define HW 512
#define KS 25
#define KH 12   // half kernel

// ---- workspace layout (float index units from base of d_ws) ----
#define WS_G      0       // [32][32]  gaussian taps (25 used)
#define WS_GLARE  1024    // [32][8]   {intensity, cx, cy, 1/rx, 1/ry}
#define WS_RS     1280    // [32]      log2(1-a)
#define WS_AMT    1312    // [32]      amount/2
#define WS_OCC    1344    // [32][4]   int: y0,y1,x0,x1
#define WS_STRK   1536    // [32][60][4] int: xc,y0,y1,active

__global__ void cp_setup_kernel(const float* __restrict__ sigma_u,
                                const float* __restrict__ glare_u,
                                const float* __restrict__ occ_u,
                                const float* __restrict__ rain_u,
                                const float* __restrict__ rain_n_u,
                                const float* __restrict__ rain_alpha_u,
                                const float* __restrict__ noise_amt_u,
                                float* __restrict__ wsF) {
    int b = threadIdx.x;
    if (b >= 32) return;
    int* wsI = (int*)wsF;

    // ---- gaussian kernel ----
    float sig = 1.0f + 3.0f * sigma_u[b];
    float inv2s2 = 1.0f / (2.0f * sig * sig);
    float g[KS];
    float sum = 0.0f;
    #pragma unroll
    for (int k = 0; k < KS; ++k) {
        float c = (float)(k - KH);
        g[k] = expf(-c * c * inv2s2);
        sum += g[k];
    }
    float inv = 1.0f / sum;
    #pragma unroll
    for (int k = 0; k < KS; ++k) wsF[WS_G + b * 32 + k] = g[k] * inv;

    // ---- glare ----
    float it = 0.4f + 0.5f * glare_u[b * 5 + 0];
    float rx = (0.1f + 0.25f * glare_u[b * 5 + 1]) * (HW * 0.5f);
    float ry = (0.1f + 0.25f * glare_u[b * 5 + 2]) * (HW * 0.5f);
    float cx = (0.2f + 0.6f * glare_u[b * 5 + 3]) * (float)HW;
    float cy = (0.2f + 0.6f * glare_u[b * 5 + 4]) * (float)HW;
    wsF[WS_GLARE + b * 8 + 0] = it;
    wsF[WS_GLARE + b * 8 + 1] = cx;
    wsF[WS_GLARE + b * 8 + 2] = cy;
    wsF[WS_GLARE + b * 8 + 3] = 1.0f / rx;
    wsF[WS_GLARE + b * 8 + 4] = 1.0f / ry;

    // ---- occlusion rect ----
    float ph = floorf((float)HW * (0.1f + 0.3f * occ_u[b * 4 + 0]));
    float pw = floorf((float)HW * (0.1f + 0.3f * occ_u[b * 4 + 1]));
    float y0 = floorf(occ_u[b * 4 + 2] * ((float)HW - ph));
    float x0 = floorf(occ_u[b * 4 + 3] * ((float)HW - pw));
    wsI[WS_OCC + b * 4 + 0] = (int)y0;
    wsI[WS_OCC + b * 4 + 1] = (int)(y0 + ph);
    wsI[WS_OCC + b * 4 + 2] = (int)x0;
    wsI[WS_OCC + b * 4 + 3] = (int)(x0 + pw);

    // ---- rain ----
    float a = 0.15f + 0.35f * rain_alpha_u[b];
    wsF[WS_RS + b] = log2f(1.0f - a);
    float n = floorf(20.0f + 41.0f * rain_n_u[b]);
    for (int s = 0; s < 60; ++s) {
        int act = ((float)s < n) ? 1 : 0;
        int xc  = (int)floorf(rain_u[(b * 60 + s) * 3 + 0] * (float)HW);
        int y0s = (int)floorf(rain_u[(b * 60 + s) * 3 + 1] * (float)(HW / 2));
        int y1s = (HW / 2) + (int)floorf(rain_u[(b * 60 + s) * 3 + 2] * (float)(HW / 2));
        wsI[WS_STRK + (b * 60 + s) * 4 + 0] = xc;
        wsI[WS_STRK + (b * 60 + s) * 4 + 1] = y0s;
        wsI[WS_STRK + (b * 60 + s) * 4 + 2] = y1s;
        wsI[WS_STRK + (b * 60 + s) * 4 + 3] = act;
    }

    // ---- noise ----
    wsF[WS_AMT + b] = (0.01f + 0.07f * noise_amt_u[b]) * 0.5f;
}

// LDS tile geometry: 64x64 output tile, 88x88 haloed input.
// A buffer padded to 96 rows (rows 88..95 zero-filled) so the WMMA feed
// needs no per-lane bounds guard -> plain aligned ds_load_b64.
#define AST 90   // A LDS row stride (floats)  -> even, keeps v2f loads 8B-aligned
#define BST 68   // B LDS row stride (floats)

__launch_bounds__(256)
__global__ void cp_fused_kernel(const float* __restrict__ x,
                                const float* __restrict__ noise,
                                const int*   __restrict__ flags,
                                const float* __restrict__ wsF,
                                float* __restrict__ out) {
    __shared__ __align__(16) float As[96 * AST];   // input tile + halo (+8 zero rows)
    __shared__ __align__(16) float Bs[96 * BST];   // horizontally blurred tile
    __shared__ int4  sStr[60];
    __shared__ int   sN;

    const int* wsI = (const int*)wsF;
    const int tid = threadIdx.x;
    const int blk = blockIdx.x;
    const int b   = blk / 192;          // 3 channels * 64 tiles
    int rem = blk - b * 192;
    const int ch  = rem / 64;
    int t = rem - ch * 64;
    const int r0 = (t >> 3) * 64;
    const int c0 = (t & 7) * 64;

    const int f0 = flags[b * 5 + 0] > 0;
    const int f1 = flags[b * 5 + 1] > 0;
    const int f2 = flags[b * 5 + 2] > 0;
    const int f3 = flags[b * 5 + 3] > 0;
    const int f4 = flags[b * 5 + 4] > 0;

    const size_t imgOff = (size_t)(b * 3 + ch) * HW * HW;
    const float* xim = x + imgOff;

    if (tid == 0) sN = 0;
    __syncthreads();

    // compact rain streaks hitting this 64-column window into LDS
    if (f3 && tid < 60) {
        const int* sp = wsI + WS_STRK + (b * 60 + tid) * 4;
        int xc = sp[0], y0 = sp[1], y1 = sp[2], act = sp[3];
        if (act && xc >= c0 && xc <= c0 + 64) {
            int p = atomicAdd(&sN, 1);
            sStr[p] = make_int4(xc, y0, y1, 0);
        }
    }

    // cooperative load of 96x88 region: 88x88 haloed input (zero-padded at the
    // image border) plus 8 all-zero rows so WMMA row-tile 5 reads zeros.
    for (int i = tid; i < 96 * 88; i += 256) {
        int ar = i / 88, ac = i - ar * 88;
        int gr = r0 - KH + ar, gc = c0 - KH + ac;
        float v = 0.0f;
        if (ar < 88 && (unsigned)gr < (unsigned)HW && (unsigned)gc < (unsigned)HW)
            v = xim[gr * HW + gc];
        As[ar * AST + ac] = v;
    }
    __syncthreads();

    const int lane  = tid & 31;
    const int wv    = tid >> 5;          // wave id 0..7 (wave32)
    const int ln16  = lane & 15;
    const int khalf = (lane >> 4) * 2;   // 0 or 2 (K sub-index of this half-wave)

    // Banded Gaussian matrix chunks G[i]: element(K=m, idx=n) = g[m-n].
    // Same per-lane layout serves as WMMA-B (horizontal) and WMMA-A (vertical).
    v2f G[10];
    #pragma unroll
    for (int i = 0; i < 10; ++i) G[i] = (v2f){0.0f, 0.0f};

    if (f0) {
        const float* gk = wsF + WS_G + b * 32;
        #pragma unroll
        for (int i = 0; i < 10; ++i) {
            int m0 = 4 * i + khalf;
            int d0 = m0 - ln16;
            int d1 = d0 + 1;
            float a0 = (d0 >= 0 && d0 < KS) ? gk[d0] : 0.0f;
            float a1 = (d1 >= 0 && d1 < KS) ? gk[d1] : 0.0f;
            G[i] = (v2f){a0, a1};
        }

        // ---- horizontal blur: 6 row-tiles x 4 col-tiles = 24 WMMA jobs, 3 per wave ----
        #pragma unroll
        for (int jj = 0; jj < 3; ++jj) {
            int job = wv + jj * 8;
            int a = job >> 2, tx = job & 3;
            int arow = a * 16 + ln16;
            int abase = arow * AST + tx * 16 + khalf;   // even -> 8B aligned
            v8f acc = {};
            #pragma unroll
            for (int i = 0; i < 10; ++i) {
                v2f av = *(const v2f*)(As + abase + 4 * i);
                acc = __builtin_amdgcn_wmma_f32_16x16x4_f32(
                          false, av, false, G[i], (short)0, acc, false, false);
            }
            int brow = a * 16 + (lane >> 4) * 8;
            int bcol = tx * 16 + ln16;
            #pragma unroll
            for (int v = 0; v < 8; ++v)
                Bs[(brow + v) * BST + bcol] = acc[v];
        }
        __syncthreads();
    }

    // per-sample epilogue params
    const float* gl = wsF + WS_GLARE + b * 8;
    const float g_it = gl[0], g_cx = gl[1], g_cy = gl[2], g_irx = gl[3], g_iry = gl[4];
    const int* oc = wsI + WS_OCC + b * 4;
    const int oy0 = oc[0], oy1 = oc[1], ox0 = oc[2], ox1 = oc[3];
    const float l2a = wsF[WS_RS + b];
    const float ah  = wsF[WS_AMT + b];

    float*       oim = out   + imgOff;
    const float* nim = noise + imgOff;

    // ---- vertical blur (2 output tiles per wave) + fused effects ----
    #pragma unroll
    for (int jj = 0; jj < 2; ++jj) {
        int job = wv * 2 + jj;
        int ty = job >> 2, tx = job & 3;
        int colL = tx * 16 + ln16;
        int X = c0 + colL;

        v8f acc = {};
        if (f0) {
            int bbase = (ty * 16 + khalf) * BST + colL;
            #pragma unroll
            for (int i = 0; i < 10; ++i) {
                v2f bv;
                bv.x = Bs[bbase + (4 * i) * BST];
                bv.y = Bs[bbase + (4 * i + 1) * BST];
                acc = __builtin_amdgcn_wmma_f32_16x16x4_f32(
                          false, G[i], false, bv, (short)0, acc, false, false);
            }
        }

        int rowB = ty * 16 + (lane >> 4) * 8;   // + v gives local row
        float dxg = ((float)X - g_cx) * g_irx;
        float dx2 = dxg * dxg;

        int cnt[8];
        #pragma unroll
        for (int v = 0; v < 8; ++v) cnt[v] = 0;
        if (f3) {
            int ns = sN;
            for (int s = 0; s < ns; ++s) {
                int4 st = sStr[s];
                if (X >= st.x - 1 && X <= st.x) {
                    #pragma unroll
                    for (int v = 0; v < 8; ++v) {
                        int Y = r0 + rowB + v;
                        cnt[v] += (Y >= st.y && Y < st.z) ? 1 : 0;
                    }
                }
            }
        }

        #pragma unroll
        for (int v = 0; v < 8; ++v) {
            int rl = rowB + v;
            int Y = r0 + rl;
            float val;
            if (f0) val = fminf(fmaxf(acc[v], 0.0f), 1.0f);
            else    val = As[(rl + KH) * AST + colL + KH];

            if (f1) {
                float dyg = ((float)Y - g_cy) * g_iry;
                val = fminf(val + g_it * __expf(-(dx2 + dyg * dyg)), 1.0f);
            }
            if (f2) {
                if (Y >= oy0 && Y < oy1 && X >= ox0 && X < ox1) val = 0.0f;
            }
            if (f3 && cnt[v] > 0) {
                float d = exp2f((float)cnt[v] * l2a);
                val = val * d + (1.0f - d);
            }
            int idx = Y * HW + X;
            if (f4) {
                float nz = nim[idx];
                val = (nz < ah) ? 0.0f : ((nz > 1.0f - ah) ? 1.0f : val);
            }
            oim[idx] = val;
        }
    }
}

extern "C" void kernel_launch(void* const* d_in, const int* in_sizes, int n_in,
                              void* d_out, int out_size, void* d_ws, size_t ws_size,
                              hipStream_t stream) {
    (void)in_sizes; (void)n_in; (void)out_size; (void)ws_size;
    const float* x            = (const float*)d_in[0];
    const float* sigma_u      = (const float*)d_in[1];
    const float* glare_u      = (const float*)d_in[2];
    const float* occ_u        = (const float*)d_in[3];
    const float* rain_u       = (const float*)d_in[4];
    const float* rain_n_u     = (const float*)d_in[5];
    const float* rain_alpha_u = (const float*)d_in[6];
    const float* noise_u      = (const float*)d_in[7];
    const float* noise_amt_u  = (const float*)d_in[8];
    const int*   apply_flags  = (const int*)d_in[9];
    float* outp = (float*)d_out;
    float* ws   = (float*)d_ws;

    cp_setup_kernel<<<1, 32, 0, stream>>>(sigma_u, glare_u, occ_u, rain_u,
                                          rain_n_u, rain_alpha_u, noise_amt_u, ws);

    // 32 samples * 3 channels * (512/64)^2 tiles = 6144 blocks
    cp_fused_kernel<<<6144, 256, 0, stream>>>(x, noise_u, apply_flags, ws, outp);
}